// GNN_61280593379663
// MI455X (gfx1250) — compile-verified
//
#include <hip/hip_runtime.h>

typedef __attribute__((ext_vector_type(2))) float v2f;
typedef __attribute__((ext_vector_type(8))) float v8f;

#define GNN_N_NODES   100000
#define GNN_N_EDGES   1600000
#define GNN_F_IN      128
#define GNN_HIDDEN    64
#define GNN_N_CLASSES 10
#define GNN_N_GRAPHS  256

// ---------------------------------------------------------------------------
// Hardware FP32 atomic add, device scope (RMW executes at L2).
// No-return form -> tracked with STOREcnt; guarantees global_atomic_add_f32
// instead of a possible compare-exchange loop from generic atomicrmw lowering.
// ---------------------------------------------------------------------------
__device__ __forceinline__ void gnn_atomic_add_f32(float* p, float v)
{
    asm volatile("global_atomic_add_f32 %0, %1, off scope:SCOPE_DEV"
                 :
                 : "v"(p), "v"(v)
                 : "memory");
}

// ---------------------------------------------------------------------------
// Dual GEMM: Yrel = act(A) @ Wrel ; Yroot = act(A) @ Wroot + bias
//   A: [M, K] row-major, Wrel/Wroot: [K, 64] row-major, outputs [M, 64].
//   Block = 128 threads = 4 waves; wave w -> 16-col tile n0 = 16*w.
//   Block b -> rows [16*b, 16*b+16). M = 100000 = 16 * 6250 exactly.
//   V_WMMA_F32_16X16X4_F32 per ISA 7.12; dual accumulators share A fragment.
// ---------------------------------------------------------------------------
template <int K, bool RELU_IN, bool HAS_BIAS>
__global__ __launch_bounds__(128) void gnn_gemm_dual(
    const float* __restrict__ A,
    const float* __restrict__ Wrel,
    const float* __restrict__ Wroot,
    const float* __restrict__ bias,
    float* __restrict__ Yrel,
    float* __restrict__ Yroot)
{
    const int lane  = threadIdx.x & 31;
    const int wave  = threadIdx.x >> 5;          // 0..3
    const int m0    = blockIdx.x * 16;
    const int n0    = wave * 16;
    const int mrow  = m0 + (lane & 15);          // A row this lane feeds
    const int ncol  = n0 + (lane & 15);          // output col this lane owns
    const int khalf = (lane >> 4) * 2;           // lanes 0-15 -> K{0,1}, 16-31 -> K{2,3}

    v8f crel  = {};
    v8f croot = {};

    const float* __restrict__ arow = A + (long)mrow * K + khalf;

#pragma unroll
    for (int k = 0; k < K; k += 4) {
        float a0 = arow[k];
        float a1 = arow[k + 1];
        if (RELU_IN) { a0 = fmaxf(a0, 0.0f); a1 = fmaxf(a1, 0.0f); }
        v2f a; a.x = a0; a.y = a1;

        const int kb = k + khalf;
        v2f brel, broot;
        brel.x  = Wrel [(kb    ) * 64 + ncol];
        brel.y  = Wrel [(kb + 1) * 64 + ncol];
        broot.x = Wroot[(kb    ) * 64 + ncol];
        broot.y = Wroot[(kb + 1) * 64 + ncol];

        crel  = __builtin_amdgcn_wmma_f32_16x16x4_f32(false, a, false, brel,
                                                      (short)0, crel,  false, false);
        croot = __builtin_amdgcn_wmma_f32_16x16x4_f32(false, a, false, broot,
                                                      (short)0, croot, false, false);
    }

    const float bv    = HAS_BIAS ? bias[ncol] : 0.0f;
    const int   rbase = m0 + ((lane >> 4) * 8);  // D: vgpr r -> row r (lanes<16) / r+8
#pragma unroll
    for (int r = 0; r < 8; ++r) {
        Yrel [(long)(rbase + r) * 64 + ncol] = crel[r];
        Yroot[(long)(rbase + r) * 64 + ncol] = croot[r] + bv;
    }
}

// ---------------------------------------------------------------------------
// Edge scatter: out[dst[e], :] += y[src[e], :]   (64 f32 per edge)
// One wave (32 lanes) per edge, two 32-wide column halves.
// y (25.6 MB) is L2-resident (192 MB L2), so gathers + atomics run at L2 rate.
// ---------------------------------------------------------------------------
__global__ __launch_bounds__(256) void gnn_edge_scatter(
    const float* __restrict__ y,
    const int* __restrict__ src,
    const int* __restrict__ dst,
    float* __restrict__ out,
    int nE)
{
    const int e    = (int)((blockIdx.x * (unsigned)blockDim.x + threadIdx.x) >> 5);
    const int lane = threadIdx.x & 31;
    if (e >= nE) return;

    const long s = src[e];
    const long d = dst[e];
    const float v0 = y[s * 64 + lane];
    const float v1 = y[s * 64 + 32 + lane];
    gnn_atomic_add_f32(&out[d * 64 + lane],      v0);
    gnn_atomic_add_f32(&out[d * 64 + 32 + lane], v1);
}

// ---------------------------------------------------------------------------
// Mean-pool accumulation: sums[batch[n], :] += h[n, :]; counts[batch[n]] += 1
// One wave per node.
// ---------------------------------------------------------------------------
__global__ __launch_bounds__(256) void gnn_pool_accum(
    const float* __restrict__ h,
    const int* __restrict__ batch,
    float* __restrict__ sums,
    float* __restrict__ counts,
    int nNodes)
{
    const int node = (int)((blockIdx.x * (unsigned)blockDim.x + threadIdx.x) >> 5);
    const int lane = threadIdx.x & 31;
    if (node >= nNodes) return;

    const int g = batch[node];
    gnn_atomic_add_f32(&sums[g * 64 + lane],      h[(long)node * 64 + lane]);
    gnn_atomic_add_f32(&sums[g * 64 + 32 + lane], h[(long)node * 64 + 32 + lane]);
    if (lane == 0)
        gnn_atomic_add_f32(&counts[g], 1.0f);
}

__global__ void gnn_zero_f32(float* __restrict__ p, int n)
{
    const int i = blockIdx.x * blockDim.x + threadIdx.x;
    if (i < n) p[i] = 0.0f;
}

// ---------------------------------------------------------------------------
// Final: out[g, c] = (sums[g,:] / max(counts[g],1)) @ W_lin[:, c] + b_lin[c]
// Block per graph (64 threads).
// ---------------------------------------------------------------------------
__global__ __launch_bounds__(64) void gnn_final_linear(
    const float* __restrict__ sums,
    const float* __restrict__ counts,
    const float* __restrict__ Wlin,   // [64, 10]
    const float* __restrict__ blin,   // [10]
    float* __restrict__ out)          // [256, 10]
{
    __shared__ float p[GNN_HIDDEN];
    const int g = blockIdx.x;
    const int t = threadIdx.x;
    const float c = fmaxf(counts[g], 1.0f);
    p[t] = sums[g * 64 + t] / c;
    __syncthreads();
    if (t < GNN_N_CLASSES) {
        float acc = blin[t];
#pragma unroll
        for (int k = 0; k < GNN_HIDDEN; ++k)
            acc = fmaf(p[k], Wlin[k * GNN_N_CLASSES + t], acc);
        out[g * GNN_N_CLASSES + t] = acc;
    }
}

// ---------------------------------------------------------------------------
extern "C" void kernel_launch(void* const* d_in, const int* in_sizes, int n_in,
                              void* d_out, int out_size, void* d_ws, size_t ws_size,
                              hipStream_t stream)
{
    (void)in_sizes; (void)n_in; (void)out_size; (void)ws_size;

    const float* x       = (const float*)d_in[0];           // [100000, 128]
    const int*   eidx    = (const int*)  d_in[1];           // [2, 1600000]
    const int*   batch   = (const int*)  d_in[2];           // [100000]
    const float* W1_rel  = (const float*)d_in[3];
    const float* b1_rel  = (const float*)d_in[4];
    const float* W1_root = (const float*)d_in[5];
    const float* W2_rel  = (const float*)d_in[6];
    const float* b2_rel  = (const float*)d_in[7];
    const float* W2_root = (const float*)d_in[8];
    const float* W3_rel  = (const float*)d_in[9];
    const float* b3_rel  = (const float*)d_in[10];
    const float* W3_root = (const float*)d_in[11];
    const float* W_lin   = (const float*)d_in[12];
    const float* b_lin   = (const float*)d_in[13];
    float*       out     = (float*)d_out;

    const int* src = eidx;
    const int* dst = eidx + GNN_N_EDGES;

    // Workspace layout (floats)
    float* ws     = (float*)d_ws;
    const long NF = (long)GNN_N_NODES * GNN_HIDDEN;         // 6.4M floats
    float* y      = ws;                                     // rel-path edge features
    float* hA     = ws + NF;                                // layer1 out / layer3 out
    float* hB     = ws + 2 * NF;                            // layer2 out
    float* sums   = ws + 3 * NF;                            // [256, 64]
    float* counts = sums + GNN_N_GRAPHS * GNN_HIDDEN;       // [256]

    const dim3 gemmGrid(GNN_N_NODES / 16), gemmBlk(128);
    const dim3 edgeGrid((GNN_N_EDGES * 32 + 255) / 256), edgeBlk(256);
    const dim3 poolGrid((GNN_N_NODES * 32 + 255) / 256), poolBlk(256);

    // ---- Layer 1: h1 = scatter(x @ W1_rel) + x @ W1_root + b1 --------------
    gnn_gemm_dual<GNN_F_IN, false, true><<<gemmGrid, gemmBlk, 0, stream>>>(
        x, W1_rel, W1_root, b1_rel, y, hA);
    gnn_edge_scatter<<<edgeGrid, edgeBlk, 0, stream>>>(y, src, dst, hA, GNN_N_EDGES);

    // ---- Layer 2 (ReLU fused into A-load): h2 ------------------------------
    gnn_gemm_dual<GNN_HIDDEN, true, true><<<gemmGrid, gemmBlk, 0, stream>>>(
        hA, W2_rel, W2_root, b2_rel, y, hB);
    gnn_edge_scatter<<<edgeGrid, edgeBlk, 0, stream>>>(y, src, dst, hB, GNN_N_EDGES);

    // ---- Layer 3 (ReLU fused): h3 -> hA ------------------------------------
    gnn_gemm_dual<GNN_HIDDEN, true, true><<<gemmGrid, gemmBlk, 0, stream>>>(
        hB, W3_rel, W3_root, b3_rel, y, hA);
    gnn_edge_scatter<<<edgeGrid, edgeBlk, 0, stream>>>(y, src, dst, hA, GNN_N_EDGES);

    // ---- Global mean pool + classifier -------------------------------------
    const int nzero = GNN_N_GRAPHS * GNN_HIDDEN + GNN_N_GRAPHS;
    gnn_zero_f32<<<(nzero + 255) / 256, 256, 0, stream>>>(sums, nzero);
    gnn_pool_accum<<<poolGrid, poolBlk, 0, stream>>>(hA, batch, sums, counts, GNN_N_NODES);
    gnn_final_linear<<<GNN_N_GRAPHS, 64, 0, stream>>>(sums, counts, W_lin, b_lin, out);
}